// GraphEmbedder_54228257080067
// MI455X (gfx1250) — compile-verified
//
#include <hip/hip_runtime.h>
#include <stdint.h>

typedef __attribute__((ext_vector_type(2))) float v2f;
typedef __attribute__((ext_vector_type(8))) float v8f;

#define FIN 128
#define FOUT 64

// ---------------------------------------------------------------------------
// 1) h = x @ W via V_WMMA_F32_16X16X4_F32 (wave32; one wave = 16x16 tile)
// ---------------------------------------------------------------------------
__global__ __launch_bounds__(256) void gemm_wmma(const float* __restrict__ x,
                                                 const float* __restrict__ W,
                                                 float* __restrict__ h, int N) {
  const int lane = threadIdx.x & 31;
  const int wave = threadIdx.x >> 5;
  const int rowHalf = wave >> 2;   // 2 row-tiles per block
  const int colTile = wave & 3;    // 4 col-tiles (64 cols)
  const int tileM = blockIdx.x * 32 + rowHalf * 16;
  const int colBase = colTile * 16;
  const int r = lane & 15;   // A: row M ; B/C/D: col N
  const int hf = lane >> 4;  // selects K (A/B) or M-half (C/D)

  int arow = tileM + r;
  if (arow >= N) arow = N - 1;  // clamp (stores are tile-guarded)
  const float* __restrict__ xrow = x + (size_t)arow * FIN;

  v8f c = {};
#pragma unroll 8
  for (int kb = 0; kb < FIN; kb += 4) {
    const int k0 = kb + 2 * hf;
    v2f a, bm;
    a[0] = xrow[k0];
    a[1] = xrow[k0 + 1];
    bm[0] = W[(size_t)k0 * FOUT + colBase + r];
    bm[1] = W[(size_t)(k0 + 1) * FOUT + colBase + r];
    c = __builtin_amdgcn_wmma_f32_16x16x4_f32(false, a, false, bm,
                                              (short)0, c, false, false);
  }

  float* __restrict__ hp = h + (size_t)(tileM + 8 * hf) * FOUT + colBase + r;
  if (tileM + 16 <= N) {
    // full tile (always true when N % 32 == 0): straight store burst
#pragma unroll
    for (int v = 0; v < 8; ++v) hp[(size_t)v * FOUT] = c[v];
  } else {
#pragma unroll
    for (int v = 0; v < 8; ++v) {
      if (tileM + 8 * hf + v < N) hp[(size_t)v * FOUT] = c[v];
    }
  }
}

// ---------------------------------------------------------------------------
// 2) degree / symmetric normalization
// ---------------------------------------------------------------------------
__global__ __launch_bounds__(256) void init_deg(float* __restrict__ deg, int N) {
  int i = blockIdx.x * 256 + threadIdx.x;
  if (i < N) deg[i] = 1.0f;  // self-loop weight
}

__global__ __launch_bounds__(256) void deg_scatter(const int* __restrict__ dst,
                                                   const float* __restrict__ ew,
                                                   float* __restrict__ deg, int E) {
  int e = blockIdx.x * 256 + threadIdx.x;
  if (e < E) unsafeAtomicAdd(&deg[dst[e]], ew[e]);
}

__global__ __launch_bounds__(256) void make_dinv(float* __restrict__ deg, int N) {
  int i = blockIdx.x * 256 + threadIdx.x;
  if (i < N) {
    float d = deg[i];
    deg[i] = d > 0.0f ? rsqrtf(d) : 0.0f;
  }
}

// acc[i,f] = b[f] + dinv[i]^2 * h[i,f]   (bias + self-loop message)
__global__ __launch_bounds__(256) void init_acc(const float* __restrict__ h,
                                                const float* __restrict__ dinv,
                                                const float* __restrict__ b,
                                                float* __restrict__ acc, int N) {
  size_t tid = (size_t)blockIdx.x * 256 + threadIdx.x;
  if (tid >= (size_t)N * FOUT) return;
  int node = (int)(tid >> 6);
  int f = (int)(tid & 63);
  float di = dinv[node];
  acc[tid] = b[f] + di * di * h[tid];
}

// ---------------------------------------------------------------------------
// 3) edge scatter: one wave per edge, 2 features per lane (L2-resident h/acc)
// ---------------------------------------------------------------------------
__global__ __launch_bounds__(256) void edge_scatter(const int* __restrict__ src,
                                                    const int* __restrict__ dst,
                                                    const float* __restrict__ ew,
                                                    const float* __restrict__ dinv,
                                                    const float* __restrict__ h,
                                                    float* __restrict__ acc, int E) {
  long long tid = (long long)blockIdx.x * 256 + threadIdx.x;
  int e = (int)(tid >> 5);
  if (e >= E) return;
  int fp = (int)(tid & 31);
  int s = src[e];
  int d = dst[e];
  float nrm = dinv[s] * ew[e] * dinv[d];
  const float* hp = h + (size_t)s * FOUT + fp * 2;
  float* ap = acc + (size_t)d * FOUT + fp * 2;
  unsafeAtomicAdd(ap, nrm * hp[0]);
  unsafeAtomicAdd(ap + 1, nrm * hp[1]);
}

// ---------------------------------------------------------------------------
// 4) attention norm, relu + score + sort keys
// ---------------------------------------------------------------------------
__global__ void attn_norm(const float* __restrict__ attn, float* __restrict__ invn,
                          int n) {
  __shared__ float red[64];
  int t = threadIdx.x;
  float v = t < n ? attn[t] : 0.0f;
  red[t] = v * v;
  __syncthreads();
  for (int s = 32; s > 0; s >>= 1) {
    if (t < s) red[t] += red[t + s];
    __syncthreads();
  }
  if (t == 0) invn[0] = rsqrtf(red[0]);
}

__device__ inline unsigned f2ord(float f) {
  unsigned u = __float_as_uint(f);
  return (u & 0x80000000u) ? ~u : (u | 0x80000000u);  // ascending-order map
}

// one wave per node: relu acc in place, score = tanh(dot(acc, attn) * invn)
__global__ __launch_bounds__(256) void score_relu(float* __restrict__ acc,
                                                  const float* __restrict__ attn,
                                                  const float* __restrict__ invn,
                                                  float* __restrict__ score,
                                                  unsigned long long* __restrict__ keys,
                                                  int N) {
  int node = blockIdx.x * 8 + (threadIdx.x >> 5);
  if (node >= N) return;
  int lane = threadIdx.x & 31;
  float* ap = acc + (size_t)node * FOUT + lane * 2;
  float v0 = ap[0], v1 = ap[1];
  v0 = v0 > 0.0f ? v0 : 0.0f;
  v1 = v1 > 0.0f ? v1 : 0.0f;
  ap[0] = v0;
  ap[1] = v1;
  float part = v0 * attn[lane * 2] + v1 * attn[lane * 2 + 1];
  for (int off = 16; off > 0; off >>= 1) part += __shfl_xor(part, off, 32);
  if (lane == 0) {
    float s = tanhf(part * invn[0]);
    score[node] = s;
    // descending score, ascending index tiebreak, ascending u64 sort
    keys[node] = ((unsigned long long)(~f2ord(s)) << 32) | (unsigned)node;
  }
}

__global__ __launch_bounds__(256) void pad_keys(unsigned long long* __restrict__ keys,
                                                int N, int SZ) {
  int i = N + blockIdx.x * 256 + threadIdx.x;
  if (i < SZ) keys[i] = 0xFFFFFFFFFFFFFFFFull;
}

__global__ __launch_bounds__(256) void bitonic_pass(unsigned long long* __restrict__ keys,
                                                    unsigned j, unsigned k) {
  unsigned i = blockIdx.x * 256 + threadIdx.x;
  unsigned ixj = i ^ j;
  if (ixj <= i) return;
  unsigned long long a = keys[i];
  unsigned long long b = keys[ixj];
  bool asc = ((i & k) == 0);
  if ((a > b) == asc) {
    keys[i] = b;
    keys[ixj] = a;
  }
}

// out[r,:] = acc[perm[r],:] * score[perm[r]]
__global__ __launch_bounds__(256) void gather_out(const float* __restrict__ acc,
                                                  const float* __restrict__ score,
                                                  const unsigned long long* __restrict__ keys,
                                                  float* __restrict__ out, int N) {
  size_t tid = (size_t)blockIdx.x * 256 + threadIdx.x;
  if (tid >= (size_t)N * FOUT) return;
  int r = (int)(tid >> 6);
  int f = (int)(tid & 63);
  unsigned idx = (unsigned)(keys[r] & 0xFFFFFFFFull);
  out[tid] = acc[(size_t)idx * FOUT + f] * score[idx];
}

// ---------------------------------------------------------------------------
extern "C" void kernel_launch(void* const* d_in, const int* in_sizes, int n_in,
                              void* d_out, int out_size, void* d_ws, size_t ws_size,
                              hipStream_t stream) {
  const float* x    = (const float*)d_in[0];
  const int*   el   = (const int*)d_in[1];   // [2, E] src row then dst row
  const float* ew   = (const float*)d_in[2];
  // d_in[3] = batch (unused, all zeros)
  const float* W    = (const float*)d_in[4];
  const float* b    = (const float*)d_in[5];
  const float* attn = (const float*)d_in[6];
  float* out = (float*)d_out;

  const int N = in_sizes[3];
  const int E = in_sizes[2];
  const int* src = el;
  const int* dst = el + E;

  int SZ = 1;
  while (SZ < N) SZ <<= 1;  // 131072 for N=100000

  // workspace layout (all 8-byte aligned for these sizes)
  char* ws = (char*)d_ws;
  const size_t nf = (size_t)N * FOUT;
  float* h    = (float*)ws;                               // nf floats
  float* acc  = (float*)(ws + nf * 4);                    // nf floats
  float* dinv = (float*)(ws + nf * 8);                    // N floats (deg->dinv)
  float* scr  = (float*)(ws + nf * 8 + (size_t)N * 4);    // N floats
  unsigned long long* keys =
      (unsigned long long*)(ws + nf * 8 + (size_t)N * 8); // SZ u64
  float* invn = (float*)(ws + nf * 8 + (size_t)N * 8 + (size_t)SZ * 8);

  // 1) GEMM (WMMA fp32)
  gemm_wmma<<<(N + 31) / 32, 256, 0, stream>>>(x, W, h, N);

  // 2) degrees -> dinv
  init_deg<<<(N + 255) / 256, 256, 0, stream>>>(dinv, N);
  deg_scatter<<<(E + 255) / 256, 256, 0, stream>>>(dst, ew, dinv, E);
  make_dinv<<<(N + 255) / 256, 256, 0, stream>>>(dinv, N);

  // 3) accumulate messages
  init_acc<<<(int)((nf + 255) / 256), 256, 0, stream>>>(h, dinv, b, acc, N);
  long long sc_threads = (long long)E * 32;
  edge_scatter<<<(int)((sc_threads + 255) / 256), 256, 0, stream>>>(
      src, dst, ew, dinv, h, acc, E);

  // 4) relu + scores + keys
  attn_norm<<<1, 64, 0, stream>>>(attn, invn, in_sizes[5]);
  score_relu<<<(N + 7) / 8, 256, 0, stream>>>(acc, attn, invn, scr, keys, N);
  pad_keys<<<((SZ - N) + 255) / 256, 256, 0, stream>>>(keys, N, SZ);

  // 5) bitonic sort (ascending u64 == descending score, stable tiebreak)
  const int sblocks = SZ / 256;
  for (unsigned k = 2; k <= (unsigned)SZ; k <<= 1)
    for (unsigned j = k >> 1; j > 0; j >>= 1)
      bitonic_pass<<<sblocks, 256, 0, stream>>>(keys, j, k);

  // 6) permuted scaled gather
  gather_out<<<(int)((nf + 255) / 256), 256, 0, stream>>>(acc, scr, keys, out, N);
}